// SageNet_6270652252188
// MI455X (gfx1250) — compile-verified
//
#include <hip/hip_runtime.h>

typedef __attribute__((ext_vector_type(2))) float v2f;
typedef __attribute__((ext_vector_type(8))) float v8f;

static constexpr int NN = 100000;   // nodes
static constexpr int NE = 600000;   // edges

static inline int ceil_div(int a, int b) { return (a + b - 1) / b; }

// ---------------- elementwise helpers ----------------

__global__ void zero_f32(float* __restrict__ p, int n) {
  int i = blockIdx.x * blockDim.x + threadIdx.x;
  if (i < n) p[i] = 0.0f;
}

__global__ void degree_kernel(const int* __restrict__ dst, float* __restrict__ deg, int E) {
  int e = blockIdx.x * blockDim.x + threadIdx.x;
  if (e < E) atomicAdd(&deg[dst[e]], 1.0f);
}

// One thread per (edge, 4-feature group): gather float4 from x[src], atomic-add into agg[dst].
// Whole layer-0 working set (~110 MB) is resident in the 192 MB L2, so these are L2 atomics.
__global__ void scatter_kernel(const float* __restrict__ x, const int* __restrict__ src,
                               const int* __restrict__ dst, float* __restrict__ agg,
                               int E, int F) {
  int t = blockIdx.x * blockDim.x + threadIdx.x;
  int groups = F >> 2;
  int e = t / groups;
  if (e >= E) return;
  int g = t - e * groups;
  int s = src[e], d = dst[e];
  const float4 v = *reinterpret_cast<const float4*>(x + (size_t)s * F + (g << 2));
  float* a = agg + (size_t)d * F + (g << 2);
  atomicAdd(a + 0, v.x);
  atomicAdd(a + 1, v.y);
  atomicAdd(a + 2, v.z);
  atomicAdd(a + 3, v.w);
}

__global__ void normalize_kernel(float* __restrict__ agg, const float* __restrict__ deg,
                                 int N, int F) {
  int i = blockIdx.x * blockDim.x + threadIdx.x;
  if (i >= N * F) return;
  float d = deg[i / F];
  agg[i] = agg[i] / fmaxf(d, 1.0f);
}

// ---------------- fused SAGE linear via WMMA f32 16x16x4 ----------------
// out = [aggn | x](N x 2*Fin) @ [[wl],[wr]](2*Fin x Fout) + b, optional ReLU.
// grid.x covers 16-row node tiles (8 waves/block, one tile each),
// grid.y selects the 16-column output tile; the block stages its 16 columns of the
// concatenated weight matrix into LDS (pair-interleaved float2, zero-padded), so the
// fully-unrolled K loop is branch-free: global_load_b64 (A) + ds_load_b64 (B) + v_wmma.
// EXEC is all-ones at every WMMA (only wave-uniform control flow before the stores).
template <int Fin, int Fout, bool RELU>
__global__ void sage_gemm_wmma(const float* __restrict__ x,
                               const float* __restrict__ aggn,
                               const float* __restrict__ wl,
                               const float* __restrict__ wr,
                               const float* __restrict__ bias,
                               float* __restrict__ out,
                               int rowtiles) {
  constexpr int KTOT = 2 * Fin;            // concatenated K
  constexpr int KP   = Fin;                // number of K-row pairs
  __shared__ float2 Bs[KP * 16];           // Bs[p*16+n] = {B[2p][col], B[2p+1][col]}
  __shared__ float  bias_s[16];

  const int ncol0 = blockIdx.y * 16;       // first global output column of this block

  for (int i = threadIdx.x; i < KP * 16; i += blockDim.x) {
    int p = i >> 4;                        // K-pair index (rows 2p, 2p+1; same side, Fin even)
    int n = i & 15;
    int col = ncol0 + n;
    float v0 = 0.0f, v1 = 0.0f;
    if (col < Fout) {
      int k0 = 2 * p;
      if (k0 < Fin) {
        v0 = wl[(size_t)k0 * Fout + col];
        v1 = wl[(size_t)(k0 + 1) * Fout + col];
      } else {
        v0 = wr[(size_t)(k0 - Fin) * Fout + col];
        v1 = wr[(size_t)(k0 - Fin + 1) * Fout + col];
      }
    }
    Bs[i] = make_float2(v0, v1);
  }
  if (threadIdx.x < 16) {
    int col = ncol0 + threadIdx.x;
    bias_s[threadIdx.x] = (col < Fout) ? bias[col] : 0.0f;
  }
  __syncthreads();

  const int rowtile = blockIdx.x * (blockDim.x >> 5) + (threadIdx.x >> 5);
  if (rowtile >= rowtiles) return;         // wave-uniform, after the barrier

  const int lane = threadIdx.x & 31;
  const int half = lane >> 4;              // fragment half: K +{0,1} vs +{2,3}
  const int idx  = lane & 15;              // A row / B,D column within tile
  const int colB = ncol0 + idx;

  const float2* arow_agg =
      reinterpret_cast<const float2*>(aggn + (size_t)(rowtile * 16 + idx) * Fin);
  const float2* arow_x =
      reinterpret_cast<const float2*>(x + (size_t)(rowtile * 16 + idx) * Fin);

  v8f acc = {};
#pragma unroll
  for (int k = 0; k < KTOT; k += 4) {
    float2 af = (k < Fin) ? arow_agg[(k >> 1) + half]
                          : arow_x[((k - Fin) >> 1) + half];
    float2 bf = Bs[((k >> 1) + half) * 16 + idx];
    v2f av = {af.x, af.y};
    v2f bv = {bf.x, bf.y};
    // (neg_a, A, neg_b, B, c_mod, C, reuse_a, reuse_b)
    acc = __builtin_amdgcn_wmma_f32_16x16x4_f32(false, av, false, bv, (short)0, acc,
                                                false, false);
  }

  // D layout: lane holds column `idx`; acc[v] is row v + half*8
  const float bcol = bias_s[idx];
  constexpr bool full = ((Fout & 15) == 0);
#pragma unroll
  for (int v = 0; v < 8; ++v) {
    float r = acc[v] + bcol;
    if (RELU) r = fmaxf(r, 0.0f);
    if (full || colB < Fout)
      out[(size_t)(rowtile * 16 + v + half * 8) * Fout + colB] = r;
  }
}

// ---------------- per-layer launchers ----------------

template <int Fin, int Fout, bool RELU>
static void launch_gemm(const float* x, const float* agg, const float* wl,
                        const float* wr, const float* b, float* out,
                        hipStream_t stream) {
  constexpr int coltiles = (Fout + 15) / 16;
  const int rowtiles = NN / 16;                      // 6250 exactly
  dim3 grid((unsigned)ceil_div(rowtiles, 8), (unsigned)coltiles, 1);
  sage_gemm_wmma<Fin, Fout, RELU><<<grid, 256, 0, stream>>>(x, agg, wl, wr, b, out,
                                                            rowtiles);
}

static void run_aggregation(const float* xin, const int* src, const int* dst,
                            const float* deg, float* agg, int Fin, hipStream_t stream) {
  const int T = 256;
  int nagg = NN * Fin;
  zero_f32<<<ceil_div(nagg, T), T, 0, stream>>>(agg, nagg);
  scatter_kernel<<<ceil_div(NE * (Fin >> 2), T), T, 0, stream>>>(xin, src, dst, agg, NE, Fin);
  normalize_kernel<<<ceil_div(nagg, T), T, 0, stream>>>(agg, deg, NN, Fin);
}

// ---------------- driver ----------------

extern "C" void kernel_launch(void* const* d_in, const int* in_sizes, int n_in,
                              void* d_out, int out_size, void* d_ws, size_t ws_size,
                              hipStream_t stream) {
  (void)in_sizes; (void)n_in; (void)out_size; (void)ws_size;

  const float* emb = (const float*)d_in[0];
  const int*   ei  = (const int*)d_in[1];
  const int*   src = ei;            // edge_index[0]
  const int*   dst = ei + NE;       // edge_index[1]
  // edge_attr (d_in[2]) unused by SAGEConv

  const float* WL[4] = {(const float*)d_in[3], (const float*)d_in[6],
                        (const float*)d_in[9], (const float*)d_in[12]};
  const float* WR[4] = {(const float*)d_in[4], (const float*)d_in[7],
                        (const float*)d_in[10], (const float*)d_in[13]};
  const float* B[4]  = {(const float*)d_in[5], (const float*)d_in[8],
                        (const float*)d_in[11], (const float*)d_in[14]};

  // workspace layout (bytes, 1MB-aligned regions)
  char* ws = (char*)d_ws;
  float* deg = (float*)(ws + 0);                        //  N floats      (0.4 MB)
  float* agg = (float*)(ws + (size_t)1  * 1048576);     //  N*128 floats  (51.2 MB)
  float* x1  = (float*)(ws + (size_t)52 * 1048576);     //  N*64  floats  (25.6 MB)
  float* x2  = (float*)(ws + (size_t)78 * 1048576);     //  N*32  floats  (12.8 MB)
  float* x3  = (float*)(ws + (size_t)91 * 1048576);     //  N*16  floats  (6.4 MB)

  const int T = 256;

  // degree (identical for all layers): zero + count once
  zero_f32<<<ceil_div(NN, T), T, 0, stream>>>(deg, NN);
  degree_kernel<<<ceil_div(NE, T), T, 0, stream>>>(dst, deg, NE);

  // Layer 0: 128 -> 64, ReLU
  run_aggregation(emb, src, dst, deg, agg, 128, stream);
  launch_gemm<128, 64, true>(emb, agg, WL[0], WR[0], B[0], x1, stream);

  // Layer 1: 64 -> 32, ReLU
  run_aggregation(x1, src, dst, deg, agg, 64, stream);
  launch_gemm<64, 32, true>(x1, agg, WL[1], WR[1], B[1], x2, stream);

  // Layer 2: 32 -> 16, ReLU
  run_aggregation(x2, src, dst, deg, agg, 32, stream);
  launch_gemm<32, 16, true>(x2, agg, WL[2], WR[2], B[2], x3, stream);

  // Layer 3: 16 -> 9, no ReLU
  run_aggregation(x3, src, dst, deg, agg, 16, stream);
  launch_gemm<16, 9, false>(x3, agg, WL[3], WR[3], B[3], (float*)d_out, stream);
}